// Net_33157147525944
// MI455X (gfx1250) — compile-verified
//
#include <hip/hip_runtime.h>
#include <math.h>

#define NN 100000
#define EE 1600000
#define GG 16
#define FF 128
#define EPSB 1e-5f
#define MINSCORE 0.1f
#define TOLK 1e-7f
#define NEGV -1e9f

typedef float v2f __attribute__((ext_vector_type(2)));
typedef float v8f __attribute__((ext_vector_type(8)));

// ---- monotone float<->uint encoding for atomic max on floats (incl. negatives) ----
__device__ __forceinline__ unsigned encf(float v){
  unsigned u = __float_as_uint(v);
  return (u & 0x80000000u) ? ~u : (u | 0x80000000u);
}
__device__ __forceinline__ float decf(unsigned u){
  return (u & 0x80000000u) ? __uint_as_float(u & 0x7fffffffu) : __uint_as_float(~u);
}
__device__ __forceinline__ void atomicMaxF(unsigned* p, float v){ atomicMax(p, encf(v)); }
__device__ __forceinline__ float eluf(float v){ return v > 0.f ? v : expm1f(v); }

// =================== GCN / TopK stage ===================
__global__ void k_deg(const int* __restrict__ ei, float* __restrict__ deg){
  int e = blockIdx.x * blockDim.x + threadIdx.x;
  if (e >= EE) return;
  atomicAdd(&deg[ei[EE + e]], 1.f);
}

__global__ void k_node1(const float* __restrict__ x, const float* __restrict__ gcn_w,
                        const float* __restrict__ deg, float* __restrict__ xw,
                        float* __restrict__ dinv){
  int n = blockIdx.x * blockDim.x + threadIdx.x;
  if (n >= NN) return;
  xw[n]   = x[n*2] * gcn_w[0] + x[n*2+1] * gcn_w[1];
  dinv[n] = rsqrtf(deg[n] + 1.f);
}

__global__ void k_gcn_edge(const int* __restrict__ ei, const float* __restrict__ xw,
                           const float* __restrict__ dinv, float* __restrict__ agg){
  int e = blockIdx.x * blockDim.x + threadIdx.x;
  if (e >= EE) return;
  int s = ei[e], d = ei[EE + e];
  atomicAdd(&agg[d], xw[s] * dinv[s] * dinv[d]);
}

__global__ void k_sraw(const float* __restrict__ agg, const float* __restrict__ xw,
                       const float* __restrict__ dinv, const float* __restrict__ gcn_b,
                       const float* __restrict__ topk_w, const int* __restrict__ batch,
                       float* __restrict__ sraw, unsigned* __restrict__ smax){
  int n = blockIdx.x * blockDim.x + threadIdx.x;
  if (n >= NN) return;
  float attn = agg[n] + xw[n] * dinv[n] * dinv[n] + gcn_b[0];
  float s = attn * topk_w[0];
  sraw[n] = s;
  atomicMaxF(&smax[batch[n]], s);
}

__global__ void k_ex(const float* __restrict__ sraw, const unsigned* __restrict__ smax,
                     const int* __restrict__ batch, float* __restrict__ score,
                     float* __restrict__ exsum){
  int n = blockIdx.x * blockDim.x + threadIdx.x;
  if (n >= NN) return;
  int b = batch[n];
  float ex = expf(sraw[n] - decf(smax[b]));
  score[n] = ex;
  atomicAdd(&exsum[b], ex);
}

__global__ void k_score(const float* __restrict__ exsum, const int* __restrict__ batch,
                        float* __restrict__ score, unsigned* __restrict__ scmax){
  int n = blockIdx.x * blockDim.x + threadIdx.x;
  if (n >= NN) return;
  int b = batch[n];
  float s = score[n] / exsum[b];
  score[n] = s;
  atomicMaxF(&scmax[b], s);
}

__global__ void k_mask(const float* __restrict__ x, const float* __restrict__ score,
                       const unsigned* __restrict__ scmax, const int* __restrict__ batch,
                       float* __restrict__ fm, float* __restrict__ h2){
  int n = blockIdx.x * blockDim.x + threadIdx.x;
  if (n >= NN) return;
  float thr = fminf(decf(scmax[batch[n]]) - TOLK, MINSCORE);
  float m = score[n] > thr ? 1.f : 0.f;
  fm[n] = m;
  float sc = score[n] * m;
  h2[n*2]   = x[n*2]   * sc;
  h2[n*2+1] = x[n*2+1] * sc;
}

// =================== filtered graph norm ===================
__global__ void k_degf(const int* __restrict__ ei, const float* __restrict__ fm,
                       float* __restrict__ degf){
  int e = blockIdx.x * blockDim.x + threadIdx.x;
  if (e >= EE) return;
  int s = ei[e], d = ei[EE + e];
  float em = fm[s] * fm[d];
  if (em != 0.f) atomicAdd(&degf[d], em);
}

__global__ void k_dinvf(const float* __restrict__ degf, float* __restrict__ dinvf){
  int n = blockIdx.x * blockDim.x + threadIdx.x;
  if (n >= NN) return;
  float dg = degf[n];
  dinvf[n] = dg > 0.f ? rsqrtf(fmaxf(dg, 1.f)) : 0.f;
}

__global__ void k_normf(const int* __restrict__ ei, const float* __restrict__ fm,
                        const float* __restrict__ dinvf, float* __restrict__ normf){
  int e = blockIdx.x * blockDim.x + threadIdx.x;
  if (e >= EE) return;
  int s = ei[e], d = ei[EE + e];
  normf[e] = dinvf[s] * dinvf[d] * fm[s] * fm[d];
}

// =================== Cheb layer 0 (2 -> 128) ===================
__global__ void k_scatter2(const int* __restrict__ ei, const float* __restrict__ normf,
                           const float* __restrict__ h2, float* __restrict__ t2){
  int e = blockIdx.x * blockDim.x + threadIdx.x;
  if (e >= EE) return;
  float nf = normf[e];
  if (nf == 0.f) return;
  int s = ei[e], d = ei[EE + e];
  atomicAdd(&t2[d*2],   -h2[s*2]   * nf);
  atomicAdd(&t2[d*2+1], -h2[s*2+1] * nf);
}

__global__ void k_cheb0(const float* __restrict__ h2, const float* __restrict__ t2,
                        const float* __restrict__ w0, const float* __restrict__ w1,
                        const float* __restrict__ bias, const float* __restrict__ bng,
                        const float* __restrict__ bnb, const float* __restrict__ bnm,
                        const float* __restrict__ bnv, const float* __restrict__ fm,
                        const int* __restrict__ batch, unsigned* __restrict__ pool,
                        float* __restrict__ H){
  int t = blockIdx.x * blockDim.x + threadIdx.x;
  if (t >= NN * FF) return;
  int n = t >> 7, f = t & 127;
  float v = h2[n*2] * w0[f] + h2[n*2+1] * w0[FF + f]
          + t2[n*2] * w1[f] + t2[n*2+1] * w1[FF + f] + bias[f];
  v = eluf(v);
  v = (v - bnm[f]) * (bng[f] * rsqrtf(bnv[f] + EPSB)) + bnb[f];
  H[n*FF + f] = v;
  float pv = v + (fm[n] > 0.f ? 0.f : NEGV);
  atomicMaxF(&pool[batch[n]*FF + f], pv);
}

// =================== 128-wide edge scatter (tx1 = -sum h[src]*normf) ===================
__global__ void k_scatter128(const int* __restrict__ ei, const float* __restrict__ normf,
                             const float* __restrict__ H, float* __restrict__ T){
  int tid = blockIdx.x * blockDim.x + threadIdx.x;
  int e = tid >> 5;
  if (e >= EE) return;
  float nf = normf[e];
  if (nf == 0.f) return;
  int c = (tid & 31) << 2;
  int s = ei[e], d = ei[EE + e];
  const float4 v = *reinterpret_cast<const float4*>(H + s*FF + c);
  float* o = T + d*FF + c;
  atomicAdd(o+0, -v.x * nf);
  atomicAdd(o+1, -v.y * nf);
  atomicAdd(o+2, -v.z * nf);
  atomicAdd(o+3, -v.w * nf);
}

// =================== Cheb 128x128 GEMM via V_WMMA_F32_16X16X4_F32 ===================
// Block = 16 nodes x 128 outputs; 8 waves, each wave owns one 16x16 tile.
// out = H@w0 + T@w1 + b, then fused ELU + BN + (masked) global-max-pool, H updated in place.
__global__ __launch_bounds__(256)
void k_cheb_gemm(const float* __restrict__ H, const float* __restrict__ T,
                 const float* __restrict__ w0, const float* __restrict__ w1,
                 const float* __restrict__ bias, const float* __restrict__ bng,
                 const float* __restrict__ bnb, const float* __restrict__ bnm,
                 const float* __restrict__ bnv, const float* __restrict__ fm,
                 const int* __restrict__ batch, unsigned* __restrict__ pool,
                 float* __restrict__ Hout){
  __shared__ float sh[16][132];   // pad 132 to break bank conflicts on column reads
  __shared__ float st[16][132];
  const int node0 = blockIdx.x * 16;
  for (int idx = threadIdx.x; idx < 16 * FF; idx += 256){
    int r = idx >> 7, c = idx & 127;
    int n = node0 + r;
    float hv = 0.f, tv = 0.f;
    if (n < NN){ hv = H[n*FF + c]; tv = T[n*FF + c]; }
    sh[r][c] = hv; st[r][c] = tv;
  }
  __syncthreads();

  const int wid   = threadIdx.x >> 5;
  const int lane  = threadIdx.x & 31;
  const int khalf = lane >> 4;          // 0: lanes 0-15, 1: lanes 16-31
  const int mn    = lane & 15;          // M for A-frag, N for B/C-frag
  const int nf0   = wid * 16;           // this wave's output-feature tile

  v8f acc = {};
  for (int kk = 0; kk < FF; kk += 4){
    const int kb = kk + 2*khalf;
    v2f ah = { sh[mn][kb], sh[mn][kb+1] };                       // A[m][k], A[m][k+1]
    v2f at = { st[mn][kb], st[mn][kb+1] };
    v2f b0 = { w0[kb*FF + nf0 + mn], w0[(kb+1)*FF + nf0 + mn] }; // B[k][n], B[k+1][n]
    v2f b1 = { w1[kb*FF + nf0 + mn], w1[(kb+1)*FF + nf0 + mn] };
    acc = __builtin_amdgcn_wmma_f32_16x16x4_f32(false, ah, false, b0, (short)0, acc, false, false);
    acc = __builtin_amdgcn_wmma_f32_16x16x4_f32(false, at, false, b1, (short)0, acc, false, false);
  }

  const int f = nf0 + mn;
  const float bscale = bng[f] * rsqrtf(bnv[f] + EPSB);
  const float bmean = bnm[f], bbeta = bnb[f], bv = bias[f];
  #pragma unroll
  for (int r = 0; r < 8; ++r){
    int n = node0 + r + 8*khalf;        // C/D layout: vgpr r -> row r + 8*khalf
    if (n < NN){
      float v = acc[r] + bv;
      v = eluf(v);
      v = (v - bmean) * bscale + bbeta;
      Hout[n*FF + f] = v;
      float pv = v + (fm[n] > 0.f ? 0.f : NEGV);
      atomicMaxF(&pool[batch[n]*FF + f], pv);
    }
  }
}

// =================== init encoded-max arrays ===================
__global__ void k_init_enc(unsigned* __restrict__ smax, unsigned* __restrict__ scmax,
                           unsigned* __restrict__ pool){
  int t = blockIdx.x * blockDim.x + threadIdx.x;
  if (t < GG){ smax[t] = encf(-INFINITY); scmax[t] = encf(-INFINITY); }
  if (t < 3 * GG * FF) pool[t] = encf(NEGV);
}

// =================== final MLP (16 graphs), single block ===================
__global__ __launch_bounds__(256)
void k_mlp(const unsigned* __restrict__ pool,
           const float* __restrict__ l0w, const float* __restrict__ l0b,
           const float* __restrict__ l1w, const float* __restrict__ l1b,
           const float* __restrict__ lfw, const float* __restrict__ lfb,
           float* __restrict__ out){
  __shared__ float g[GG][3*FF];
  __shared__ float m1[GG][32];
  __shared__ float m2[GG][8];
  __shared__ float lg[GG][3];
  const int t = threadIdx.x;
  for (int idx = t; idx < 3*GG*FF; idx += 256){
    int i = idx >> 11;                 // pool slab (2048 entries per Cheb layer)
    int rem = idx & 2047;
    int gr = rem >> 7, f = rem & 127;
    g[gr][i*FF + f] = decf(pool[idx]);
  }
  __syncthreads();
  for (int o = t; o < GG*32; o += 256){
    int gr = o >> 5, c = o & 31;
    float s = l0b[c];
    for (int k = 0; k < 3*FF; ++k) s += g[gr][k] * l0w[k*32 + c];
    m1[gr][c] = eluf(s);
  }
  __syncthreads();
  if (t < GG*8){
    int gr = t >> 3, c = t & 7;
    float s = l1b[c];
    for (int k = 0; k < 32; ++k) s += m1[gr][k] * l1w[k*8 + c];
    m2[gr][c] = eluf(s);
  }
  __syncthreads();
  if (t < GG*3){
    int gr = t / 3, c = t % 3;
    float s = lfb[c];
    for (int k = 0; k < 8; ++k) s += m2[gr][k] * lfw[k*3 + c];
    lg[gr][c] = s;
  }
  __syncthreads();
  if (t < GG){
    float a = lg[t][0], b = lg[t][1], c = lg[t][2];
    float m = fmaxf(a, fmaxf(b, c));
    float lse = m + logf(expf(a-m) + expf(b-m) + expf(c-m));
    out[t*3+0] = a - lse;
    out[t*3+1] = b - lse;
    out[t*3+2] = c - lse;
  }
}

// =================== host launch ===================
extern "C" void kernel_launch(void* const* d_in, const int* in_sizes, int n_in,
                              void* d_out, int out_size, void* d_ws, size_t ws_size,
                              hipStream_t stream) {
  const float* x      = (const float*)d_in[0];
  const int*   ei     = (const int*)  d_in[1];
  const int*   batch  = (const int*)  d_in[2];
  const float* gcn_w  = (const float*)d_in[3];
  const float* gcn_b  = (const float*)d_in[4];
  const float* topk_w = (const float*)d_in[5];
  const float* c0w0 = (const float*)d_in[6],  *c0w1 = (const float*)d_in[7],  *c0b = (const float*)d_in[8];
  const float* c1w0 = (const float*)d_in[9],  *c1w1 = (const float*)d_in[10], *c1b = (const float*)d_in[11];
  const float* c2w0 = (const float*)d_in[12], *c2w1 = (const float*)d_in[13], *c2b = (const float*)d_in[14];
  const float* bng = (const float*)d_in[15], *bnb = (const float*)d_in[16];
  const float* bnm = (const float*)d_in[17], *bnv = (const float*)d_in[18];
  const float* l0w = (const float*)d_in[19], *l0b = (const float*)d_in[20];
  const float* l1w = (const float*)d_in[21], *l1b = (const float*)d_in[22];
  const float* lfw = (const float*)d_in[23], *lfb = (const float*)d_in[24];
  float* out = (float*)d_out;

  char* wp = (char*)d_ws;
  auto alloc = [&](size_t bytes)->char*{
    char* p = wp; wp += (bytes + 255) & ~(size_t)255; return p;
  };
  float*    H     = (float*)   alloc((size_t)NN * FF * 4);
  float*    T     = (float*)   alloc((size_t)NN * FF * 4);
  float*    normf = (float*)   alloc((size_t)EE * 4);
  float*    xw    = (float*)   alloc((size_t)NN * 4);
  float*    dinv  = (float*)   alloc((size_t)NN * 4);
  float*    deg   = (float*)   alloc((size_t)NN * 4);
  float*    agg   = (float*)   alloc((size_t)NN * 4);
  float*    sraw  = (float*)   alloc((size_t)NN * 4);
  float*    score = (float*)   alloc((size_t)NN * 4);
  float*    fm    = (float*)   alloc((size_t)NN * 4);
  float*    degf  = (float*)   alloc((size_t)NN * 4);
  float*    dinvf = (float*)   alloc((size_t)NN * 4);
  float*    h2    = (float*)   alloc((size_t)NN * 2 * 4);
  float*    t2    = (float*)   alloc((size_t)NN * 2 * 4);
  unsigned* smax  = (unsigned*)alloc(GG * 4);
  unsigned* scmax = (unsigned*)alloc(GG * 4);
  float*    exsum = (float*)   alloc(GG * 4);
  unsigned* pool  = (unsigned*)alloc(3 * GG * FF * 4);

  const int TB = 256;
  const int gridE  = (EE + TB - 1) / TB;         // 6250
  const int gridN  = (NN + TB - 1) / TB;         // 391
  const int gridNF = (NN * FF + TB - 1) / TB;    // 50000
  const int gridS  = (EE * 32 + TB - 1) / TB;    // 200000
  const int gridG  = NN / 16;                    // 6250 (N divisible by 16)

  // zero accumulators (graph-capture-safe)
  hipMemsetAsync(deg,   0, (size_t)NN * 4, stream);
  hipMemsetAsync(agg,   0, (size_t)NN * 4, stream);
  hipMemsetAsync(degf,  0, (size_t)NN * 4, stream);
  hipMemsetAsync(t2,    0, (size_t)NN * 2 * 4, stream);
  hipMemsetAsync(exsum, 0, GG * 4, stream);
  k_init_enc<<<(3*GG*FF + TB - 1)/TB, TB, 0, stream>>>(smax, scmax, pool);

  // GCN + TopK
  k_deg     <<<gridE, TB, 0, stream>>>(ei, deg);
  k_node1   <<<gridN, TB, 0, stream>>>(x, gcn_w, deg, xw, dinv);
  k_gcn_edge<<<gridE, TB, 0, stream>>>(ei, xw, dinv, agg);
  k_sraw    <<<gridN, TB, 0, stream>>>(agg, xw, dinv, gcn_b, topk_w, batch, sraw, smax);
  k_ex      <<<gridN, TB, 0, stream>>>(sraw, smax, batch, score, exsum);
  k_score   <<<gridN, TB, 0, stream>>>(exsum, batch, score, scmax);
  k_mask    <<<gridN, TB, 0, stream>>>(x, score, scmax, batch, fm, h2);

  // filtered-graph sym-norm
  k_degf  <<<gridE, TB, 0, stream>>>(ei, fm, degf);
  k_dinvf <<<gridN, TB, 0, stream>>>(degf, dinvf);
  k_normf <<<gridE, TB, 0, stream>>>(ei, fm, dinvf, normf);

  // Cheb layer 0 (2 -> 128)
  k_scatter2<<<gridE, TB, 0, stream>>>(ei, normf, h2, t2);
  k_cheb0<<<gridNF, TB, 0, stream>>>(h2, t2, c0w0, c0w1, c0b,
                                     bng, bnb, bnm, bnv, fm, batch, pool, H);

  // Cheb layer 1 (128 -> 128): scatter + WMMA GEMM (+fused ELU/BN/pool)
  hipMemsetAsync(T, 0, (size_t)NN * FF * 4, stream);
  k_scatter128<<<gridS, TB, 0, stream>>>(ei, normf, H, T);
  k_cheb_gemm<<<gridG, TB, 0, stream>>>(H, T, c1w0, c1w1, c1b,
                                        bng + FF, bnb + FF, bnm + FF, bnv + FF,
                                        fm, batch, pool + GG*FF, H);

  // Cheb layer 2
  hipMemsetAsync(T, 0, (size_t)NN * FF * 4, stream);
  k_scatter128<<<gridS, TB, 0, stream>>>(ei, normf, H, T);
  k_cheb_gemm<<<gridG, TB, 0, stream>>>(H, T, c2w0, c2w1, c2b,
                                        bng + 2*FF, bnb + 2*FF, bnm + 2*FF, bnv + 2*FF,
                                        fm, batch, pool + 2*GG*FF, H);

  // readout MLP + log_softmax
  k_mlp<<<1, TB, 0, stream>>>(pool, l0w, l0b, l1w, l1b, lfw, lfb, out);

  (void)in_sizes; (void)n_in; (void)out_size; (void)ws_size;
}